// GroundingDINOLoss_12455405158711
// MI455X (gfx1250) — compile-verified
//
#include <hip/hip_runtime.h>
#include <hip/hip_bf16.h>
#include <math.h>

#define B_ 64
#define Q_ 900
#define N_ 100
#define C_ 256
#define EPSF 1e-6f
#define BOX_W 5.0f
#define GIOU_W 2.0f
#define CLS_W 1.0f

#define NWAVE 8   // 256 threads / wave32

typedef __attribute__((ext_vector_type(2))) float v2f;
typedef __attribute__((ext_vector_type(8))) float v8f;

// ---------------------------------------------------------------------------
// Kernel 1: one workgroup per batch image.
//   - load pred boxes -> LDS as xyxy + area
//   - 100 sequential greedy argmax steps. Critical path optimization for the
//     serial chain: intra-wave argmax via __shfl_xor (ds_bpermute, no
//     barriers), then only 8 wave winners cross LDS => 2 barriers per step
//     instead of ~9 with a full 256-wide LDS tree.
//   - tie-break = (higher val) else (lower index), matching jnp.argmax-first
//   - then BOX_W*L1 + GIOU_W*(1-giou) partial for this batch
// ---------------------------------------------------------------------------
__global__ __launch_bounds__(256) void match_box_kernel(
    const float* __restrict__ pred_boxes,    // [B,Q,4] cxcywh
    const float* __restrict__ target_boxes,  // [B,N,4] xyxy
    int* __restrict__ ws_idx,                // [B,N]
    float* __restrict__ ws_box)              // [B]
{
  __shared__ float px0[Q_], py0[Q_], px1[Q_], py1[Q_], parea[Q_];
  __shared__ unsigned char used[Q_];
  __shared__ float s_wval[NWAVE];
  __shared__ int   s_widx[NWAVE];
  __shared__ int   s_match[N_];

  const int b    = blockIdx.x;
  const int tid  = threadIdx.x;
  const int lane = tid & 31;
  const int wave = tid >> 5;
  const float* pb = pred_boxes   + (size_t)b * Q_ * 4;
  const float* tb = target_boxes + (size_t)b * N_ * 4;

  for (int q = tid; q < Q_; q += 256) {
    float cx = pb[q * 4 + 0], cy = pb[q * 4 + 1];
    float w  = pb[q * 4 + 2], h  = pb[q * 4 + 3];
    float x0 = cx - 0.5f * w, y0 = cy - 0.5f * h;
    float x1 = cx + 0.5f * w, y1 = cy + 0.5f * h;
    px0[q] = x0; py0[q] = y0; px1[q] = x1; py1[q] = y1;
    parea[q] = (x1 - x0) * (y1 - y0);
    used[q] = 0;
  }
  __syncthreads();

  for (int n = 0; n < N_; ++n) {
    const float tx0 = tb[n * 4 + 0], ty0 = tb[n * 4 + 1];
    const float tx1 = tb[n * 4 + 2], ty1 = tb[n * 4 + 3];
    const float tarea = (tx1 - tx0) * (ty1 - ty0);

    float best = -INFINITY;
    int   bidx = 0x7FFFFFFF;
    for (int q = tid; q < Q_; q += 256) {
      if (used[q]) continue;
      float ltx = fmaxf(px0[q], tx0), lty = fmaxf(py0[q], ty0);
      float rbx = fminf(px1[q], tx1), rby = fminf(py1[q], ty1);
      float iw = fmaxf(rbx - ltx, 0.0f), ih = fmaxf(rby - lty, 0.0f);
      float inter = iw * ih;
      float uni   = parea[q] + tarea - inter;
      float iou   = inter / (uni + EPSF);
      float ex0 = fminf(px0[q], tx0), ey0 = fminf(py0[q], ty0);
      float ex1 = fmaxf(px1[q], tx1), ey1 = fmaxf(py1[q], ty1);
      float ew = fmaxf(ex1 - ex0, 0.0f), eh = fmaxf(ey1 - ey0, 0.0f);
      float earea = ew * eh;
      float g = iou - (earea - uni) / (earea + EPSF);
      if (g > best) { best = g; bidx = q; }   // ascending q -> lowest idx on tie
    }

    // intra-wave (val,idx) argmax, no barriers (symmetric combine -> all
    // lanes converge to the same deterministic winner)
#pragma unroll
    for (int off = 16; off > 0; off >>= 1) {
      float ov = __shfl_xor(best, off);
      int   oi = __shfl_xor(bidx, off);
      if (ov > best || (ov == best && oi < bidx)) { best = ov; bidx = oi; }
    }
    if (lane == 0) { s_wval[wave] = best; s_widx[wave] = bidx; }
    __syncthreads();
    if (tid == 0) {
      float bv = s_wval[0]; int bi = s_widx[0];
#pragma unroll
      for (int w2 = 1; w2 < NWAVE; ++w2) {
        float v2v = s_wval[w2]; int i2 = s_widx[w2];
        if (v2v > bv || (v2v == bv && i2 < bi)) { bv = v2v; bi = i2; }
      }
      s_match[n] = bi;
      used[bi] = 1;
      ws_idx[b * N_ + n] = bi;
    }
    __syncthreads();
  }

  // box losses for the matched pairs of this batch
  float contrib = 0.0f;
  if (tid < N_) {
    const int n = tid;
    const int q = s_match[n];
    float cx = pb[q * 4 + 0], cy = pb[q * 4 + 1];
    float w  = pb[q * 4 + 2], h  = pb[q * 4 + 3];
    float tx0 = tb[n * 4 + 0], ty0 = tb[n * 4 + 1];
    float tx1 = tb[n * 4 + 2], ty1 = tb[n * 4 + 3];
    float tcx = 0.5f * (tx0 + tx1), tcy = 0.5f * (ty0 + ty1);
    float tw = tx1 - tx0, th = ty1 - ty0;
    float l1 = fabsf(cx - tcx) + fabsf(cy - tcy) + fabsf(w - tw) + fabsf(h - th);

    float x0 = px0[q], y0 = py0[q], x1 = px1[q], y1 = py1[q];
    float a1 = parea[q];
    float a2 = tw * th;
    float ltx = fmaxf(x0, tx0), lty = fmaxf(y0, ty0);
    float rbx = fminf(x1, tx1), rby = fminf(y1, ty1);
    float iw = fmaxf(rbx - ltx, 0.0f), ih = fmaxf(rby - lty, 0.0f);
    float inter = iw * ih;
    float uni   = a1 + a2 - inter;
    float iou   = inter / (uni + EPSF);
    float ex0 = fminf(x0, tx0), ey0 = fminf(y0, ty0);
    float ex1 = fmaxf(x1, tx1), ey1 = fmaxf(y1, ty1);
    float ew = fmaxf(ex1 - ex0, 0.0f), eh = fmaxf(ey1 - ey0, 0.0f);
    float ea = ew * eh;
    float g = iou - (ea - uni) / (ea + EPSF);

    contrib = BOX_W * l1 + GIOU_W * (1.0f - g);
  }
  // wave sum (shfl), then 8 partials through LDS
#pragma unroll
  for (int off = 16; off > 0; off >>= 1) contrib += __shfl_xor(contrib, off);
  if (lane == 0) s_wval[wave] = contrib;
  __syncthreads();
  if (tid == 0) {
    float s = 0.0f;
#pragma unroll
    for (int w2 = 0; w2 < NWAVE; ++w2) s += s_wval[w2];
    ws_box[b] = s;
  }
}

// ---------------------------------------------------------------------------
// Kernel 2: one wave32 per matched row. Stable BCE-with-logits over C=256,
// reduced with chained V_WMMA_F32_16X16X4_F32 against an all-ones B matrix.
//   lane l holds 8 values; WMMA t consumes v[2t], v[2t+1] as A[l%16][{0,1}]
//   (lanes 16..31 supply K=2,3 of the same rows per the ISA A layout), so the
//   four chained WMMAs accumulate every element exactly once into D.
//   With B == ones: total = sum_m D[m][0] = lane(j) 8 comps + lane(j^16)'s.
// ---------------------------------------------------------------------------
__global__ __launch_bounds__(256) void cls_kernel(
    const float* __restrict__ pred_logits,   // [B,Q,C]
    const int* __restrict__ target_labels,   // [B,N]
    const int* __restrict__ ws_idx,          // [B,N]
    float* __restrict__ ws_cls)              // [B*N]
{
  const int lane = threadIdx.x & 31;
  const int wave = threadIdx.x >> 5;
  const int row  = blockIdx.x * 8 + wave;    // 0 .. B*N-1 (grid sized exactly)
  const int b    = row / N_;
  const int q    = ws_idx[row];
  const int label = target_labels[row];
  const float* src = pred_logits + ((size_t)b * Q_ + q) * C_;

  const int c0 = lane * 8;
  float4 A0 = *(const float4*)(src + c0);
  float4 A1 = *(const float4*)(src + c0 + 4);
  float x[8] = {A0.x, A0.y, A0.z, A0.w, A1.x, A1.y, A1.z, A1.w};

  float v[8];
#pragma unroll
  for (int i = 0; i < 8; ++i) {
    float xv = x[i];
    float tv = ((c0 + i) == label) ? xv : 0.0f;
    v[i] = fmaxf(xv, 0.0f) - tv + log1pf(expf(-fabsf(xv)));
  }

  v8f acc = {0.f, 0.f, 0.f, 0.f, 0.f, 0.f, 0.f, 0.f};
  v2f ones; ones[0] = 1.0f; ones[1] = 1.0f;
#pragma unroll
  for (int t = 0; t < 4; ++t) {
    v2f a; a[0] = v[2 * t]; a[1] = v[2 * t + 1];
    acc = __builtin_amdgcn_wmma_f32_16x16x4_f32(
        /*neg_a=*/false, a, /*neg_b=*/false, ones,
        /*c_mod=*/(short)0, acc, /*reuse_a=*/false, /*reuse_b=*/false);
  }

  float s = acc[0] + acc[1] + acc[2] + acc[3] + acc[4] + acc[5] + acc[6] + acc[7];
  s += __shfl_xor(s, 16);   // combine D rows 0-7 (lane j) with rows 8-15 (lane j^16)
  if (lane == 0) ws_cls[row] = s;
}

// ---------------------------------------------------------------------------
// Kernel 3: deterministic final reduction + scaling.
// ---------------------------------------------------------------------------
__global__ __launch_bounds__(256) void reduce_kernel(
    const float* __restrict__ ws_box,   // [B]
    const float* __restrict__ ws_cls,   // [B*N]
    float* __restrict__ out)
{
  __shared__ float s_wval[NWAVE];
  const int tid  = threadIdx.x;
  const int lane = tid & 31;
  const int wave = tid >> 5;
  float s = 0.0f;
  for (int i = tid; i < B_; i += 256) s += ws_box[i];
  for (int i = tid; i < B_ * N_; i += 256) s += CLS_W * ws_cls[i];
#pragma unroll
  for (int off = 16; off > 0; off >>= 1) s += __shfl_xor(s, off);
  if (lane == 0) s_wval[wave] = s;
  __syncthreads();
  if (tid == 0) {
    float t = 0.0f;
#pragma unroll
    for (int w2 = 0; w2 < NWAVE; ++w2) t += s_wval[w2];
    out[0] = t / (float)(B_ * N_);
  }
}

extern "C" void kernel_launch(void* const* d_in, const int* in_sizes, int n_in,
                              void* d_out, int out_size, void* d_ws, size_t ws_size,
                              hipStream_t stream) {
  const float* pred_boxes    = (const float*)d_in[0];  // [B,Q,4]
  const float* pred_logits   = (const float*)d_in[1];  // [B,Q,C]
  const float* target_boxes  = (const float*)d_in[2];  // [B,N,4]
  const int*   target_labels = (const int*)d_in[3];    // [B,N]

  char*  ws     = (char*)d_ws;
  int*   ws_idx = (int*)ws;                                     // B*N ints
  float* ws_box = (float*)(ws + (size_t)B_ * N_ * sizeof(int)); // B floats
  float* ws_cls = ws_box + B_;                                  // B*N floats

  match_box_kernel<<<B_, 256, 0, stream>>>(pred_boxes, target_boxes, ws_idx, ws_box);
  cls_kernel<<<(B_ * N_) / 8, 256, 0, stream>>>(pred_logits, target_labels, ws_idx, ws_cls);
  reduce_kernel<<<1, 256, 0, stream>>>(ws_box, ws_cls, (float*)d_out);
}